// GetLaplacian_249108103362
// MI455X (gfx1250) — compile-verified
//
#include <hip/hip_runtime.h>

// Normalized graph Laplacian on MI455X (gfx1250, wave32).
//   deg[b,j] = sum_i A[b,i,j]   (column sums, reference uses axis=1)
//   d        = rsqrt(deg)
//   L[b,i,j] = (i==j) - d_i * A[b,i,j] * d_j
//
// Bandwidth-bound: ~768 MiB total traffic -> ~33 us floor at 23.3 TB/s.
// Pass 1 does the column reduction with V_WMMA_F32_16X16X4_F32 using an
// all-ones A operand (ones^T @ tile == column sums); result is invariant to
// the K-lane permutation of the B layout, so only the N-in-lanes mapping
// matters (lanes 0-15 = columns, per the documented B/C/D striping).

typedef __attribute__((ext_vector_type(2))) float v2f;
typedef __attribute__((ext_vector_type(4))) float v4f;
typedef __attribute__((ext_vector_type(8))) float v8f;

#define NB 16
#define NN 2048

// ---------------- Pass 1: d = rsqrt(column sums) via WMMA ----------------
// Grid: NB * (NN/16) = 2048 blocks of 256 threads (8 waves).
// Block -> (batch b, 16-column strip). Wave w sums rows [256w, 256w+256).
__global__ __launch_bounds__(256) void colsum_rsqrt_kernel(
    const float* __restrict__ A, float* __restrict__ d)
{
    __shared__ float red[8][16];

    const int tid   = threadIdx.x;
    const int wave  = tid >> 5;          // 0..7
    const int lane  = tid & 31;
    const int strip = blockIdx.x & 127;  // 128 strips of 16 columns
    const int b     = blockIdx.x >> 7;   // batch index

    const int col    = (strip << 4) + (lane & 15);
    const int rowoff = (lane >> 4) * 2;  // lanes 16-31 take rows +2,+3 of each K-group
    const int row0   = wave * 256;

    const float* p = A + (long)b * NN * NN + (long)(row0 + rowoff) * NN + col;

    v2f ones; ones.x = 1.0f; ones.y = 1.0f;   // A-matrix = all ones (layout-agnostic)
    v8f c = {};                               // f32 16x16 accumulator (8 VGPRs)

    // 256 rows / (K=4 per WMMA) = 64 iterations. Per iter each lane supplies
    // two rows of its column into the B operand (2 VGPRs = 4x16 f32 tile).
    #pragma unroll 4
    for (int it = 0; it < 64; ++it) {
        v2f bv;
        bv.x = p[0];
        bv.y = p[NN];
        p += 4 * NN;
        // D = ones(16x4) x B(4x16) + C  => every row of D = column sums
        c = __builtin_amdgcn_wmma_f32_16x16x4_f32(
                /*neg_a=*/false, ones, /*neg_b=*/false, bv,
                /*c_mod=*/(short)0, c, /*reuse_a=*/false, /*reuse_b=*/false);
    }

    // D layout: VGPR0, lanes 0-15 hold (M=0, N=lane) = colsum of column lane.
    if (lane < 16) red[wave][lane] = c[0];
    __syncthreads();

    if (tid < 16) {
        float s = 0.0f;
        #pragma unroll
        for (int w = 0; w < 8; ++w) s += red[w][tid];
        d[b * NN + (strip << 4) + tid] = rsqrtf(s);
    }
}

// ---------------- Pass 2: L = diag-fix( -d_i * A * d_j ) -----------------
// One float4 per thread, fully coalesced. Output is write-once -> NT stores.
__global__ __launch_bounds__(256) void laplacian_kernel(
    const float* __restrict__ A, const float* __restrict__ d,
    float* __restrict__ L)
{
    const long vid = (long)blockIdx.x * 256 + threadIdx.x; // float4 index
    const int  j4  = (int)(vid & 511) << 2;                // 512 float4 per row
    const long row = vid >> 9;                             // b*2048 + i
    const int  i   = (int)(row & (NN - 1));
    const int  b   = (int)(row >> 11);

    const float di = d[b * NN + i];
    const v4f   dj = *(const v4f*)(d + b * NN + j4);

    v4f a = __builtin_nontemporal_load((const v4f*)(A + (vid << 2)));

    v4f o = (-di) * a * dj;

    const int t = i - j4;
    if ((unsigned)t < 4u) o[t] += 1.0f;   // identity on the diagonal

    __builtin_nontemporal_store(o, (v4f*)(L + (vid << 2)));
}

// ------------------------------- launch ----------------------------------
extern "C" void kernel_launch(void* const* d_in, const int* in_sizes, int n_in,
                              void* d_out, int out_size, void* d_ws, size_t ws_size,
                              hipStream_t stream)
{
    const float* A = (const float*)d_in[0];
    float*       L = (float*)d_out;
    float*       dvec = (float*)d_ws;            // NB*NN floats = 128 KiB scratch

    // Pass 1: 16 batches * 128 strips = 2048 workgroups
    colsum_rsqrt_kernel<<<NB * (NN / 16), 256, 0, stream>>>(A, dvec);

    // Pass 2: 16*2048*2048/4 float4 / 256 threads = 65536 workgroups
    laplacian_kernel<<<(int)((long)NB * NN * NN / 4 / 256), 256, 0, stream>>>(A, dvec, L);
}